// GRUFusion_1958505087030
// MI455X (gfx1250) — compile-verified
//
#include <hip/hip_runtime.h>

#define DIM 96
#define CCH 24

typedef __attribute__((ext_vector_type(4))) unsigned int v4u;
typedef __attribute__((ext_vector_type(8))) int v8i;
typedef __attribute__((ext_vector_type(4))) int v4i;

// ---------------------------------------------------------------------------
// TDM-based bulk fill. Each block fills a 64KB LDS tile with the fill value,
// then wave 0 issues one tensor_store_from_lds in ITERATE mode:
//   lds_addr_increment = 0  -> DMA re-reads the same LDS tile each iteration
//   global_addr_increment = TILE_FLOATS -> consecutive 64KB chunks in memory
// One DMA op streams iters*64KB to HBM per workgroup.
// ---------------------------------------------------------------------------
#define TDM_TILE_FLOATS 16384  // 64 KB

__global__ void tdm_fill_kernel(float* __restrict__ dst, int iters, float v) {
    __shared__ float tile[TDM_TILE_FLOATS];

    // Vectorized LDS fill: ds_store_b128 per lane-iteration.
    float4* t4 = (float4*)tile;
    const float4 vv = make_float4(v, v, v, v);
    for (int i = threadIdx.x; i < TDM_TILE_FLOATS / 4; i += blockDim.x) t4[i] = vv;

    // The TDM reads LDS through the descriptor's lds_addr field, which alias
    // analysis cannot see — escape the tile pointer + full memory clobber so
    // the fill stores are not dead-store-eliminated.
    asm volatile("" : : "r"(tile) : "memory");
    __syncthreads();

    if (threadIdx.x < 32) {  // wave 0 of the block issues the DMA
        unsigned long long gaddr =
            (unsigned long long)(uintptr_t)dst +
            (unsigned long long)blockIdx.x *
                ((unsigned long long)TDM_TILE_FLOATS * (unsigned)iters * 4ull);

        // --- D# group 0: flags / lds_addr / global_addr / type ---
        v4u g0;
        g0.x = 1u;                                   // count=1 (valid user D#)
        g0.y = 0u;                                   // lds_addr = 0 (tile is sole LDS alloc)
        g0.z = (unsigned)(gaddr & 0xffffffffull);    // global_addr[31:0]
        g0.w = (unsigned)(gaddr >> 32) | (2u << 30); // global_addr[56:32] | type=2

        // --- D# group 1 ---
        const unsigned tdim0 = 0x7fffffffu;          // tensor_dim0: huge, no OOB clip
        v8i g1;
        g1[0] = (2 << 16) | (1 << 19);               // data_size=4B, iterate_enable=1
        g1[1] = (int)((tdim0 & 0xffffu) << 16);      // [15:0] atomic_bar_addr=0, [31:16] tensor_dim0 lo
        g1[2] = (int)((tdim0 >> 16) | (1u << 16));   // [15:0] tensor_dim0 hi, [31:16] tensor_dim1 lo (=1)
        g1[3] = (int)(TDM_TILE_FLOATS << 16);        // [15:0] tensor_dim1 hi=0, [31:16] tile_dim0
        g1[4] = 0;                                   // tile_dim1=0 (unused), tile_dim2=0
        g1[5] = (int)tdim0;                          // tensor_dim0_stride lo
        g1[6] = 0;                                   // stride hi / dim1_stride lo
        g1[7] = 0;

        // --- D# group 2 (iterate mode) ---
        v4i g2;
        g2[0] = 1;                                   // tensor_dim2
        g2[1] = 0;                                   // lds_addr_increment = 0 (reuse tile)
        g2[2] = TDM_TILE_FLOATS;                     // global_addr_increment (data_size units)
        g2[3] = (int)(((unsigned)(iters - 1) & 0xffffu) << 16);  // iterate_count

        v4i g3 = {0, 0, 0, 0};
        v8i g4 = {0, 0, 0, 0, 0, 0, 0, 0};           // unused extra group (clang-23 arity)

        __builtin_amdgcn_tensor_store_from_lds(g0, g1, g2, g3, g4, 0);
        __builtin_amdgcn_s_wait_tensorcnt(0);        // LDS must stay live until DMA drains
    }
}

// updated_coords = (float)current_coords
__global__ void i2f_kernel(const int* __restrict__ in, float* __restrict__ out, int n) {
    int i = blockIdx.x * blockDim.x + threadIdx.x;
    if (i < n) out[i] = (float)in[i];
}

// ---------------------------------------------------------------------------
// Occupancy scatter: valid_volume[x,y,z] = 1 (coords guaranteed in [0,DIM)).
// ---------------------------------------------------------------------------
__global__ void scatter_occ_kernel(const int* __restrict__ coords,
                                   int* __restrict__ occ, int n) {
    int i = blockIdx.x * blockDim.x + threadIdx.x;
    if (i >= n) return;
    int x = coords[3 * i + 0];
    int y = coords[3 * i + 1];
    int z = coords[3 * i + 2];
    occ[(x * DIM + y) * DIM + z] = 1;
}

// ---------------------------------------------------------------------------
// Scatter a 24-float feature row per point: 6x global_load_b128 + 6x store_b128.
// ---------------------------------------------------------------------------
__global__ void scatter_feat_kernel(const int* __restrict__ coords,
                                    const float* __restrict__ vals,
                                    float* __restrict__ vol, int n) {
    int i = blockIdx.x * blockDim.x + threadIdx.x;
    if (i >= n) return;
    int x = coords[3 * i + 0];
    int y = coords[3 * i + 1];
    int z = coords[3 * i + 2];
    const float4* __restrict__ src = (const float4*)(vals + (long)i * CCH);
    float4* __restrict__ dst =
        (float4*)(vol + (long)((x * DIM + y) * DIM + z) * CCH);
#pragma unroll
    for (int k = 0; k < CCH / 4; ++k) dst[k] = src[k];
}

// ---------------------------------------------------------------------------
// Global pass: gc = raw - origin; valid = in-bounds && occ!=0; write flag;
// scatter 24-float rows for valid points. Prefetch the feature row early so
// the dependent occ load latency is hidden (gfx1250 global_prefetch_b8).
// ---------------------------------------------------------------------------
__global__ void global_pass_kernel(const int* __restrict__ gcr,
                                   const float* __restrict__ gvals,
                                   const int* __restrict__ origin,
                                   const int* __restrict__ occ,
                                   float* __restrict__ valid_out,
                                   float* __restrict__ gvol, int n) {
    int i = blockIdx.x * blockDim.x + threadIdx.x;
    if (i >= n) return;
    const float* row = gvals + (long)i * CCH;
    __builtin_prefetch(row, 0, 0);  // global_prefetch_b8: warm row while occ load resolves
    int x = gcr[3 * i + 0] - origin[0];
    int y = gcr[3 * i + 1] - origin[1];
    int z = gcr[3 * i + 2] - origin[2];
    bool inb = ((unsigned)x < DIM) && ((unsigned)y < DIM) && ((unsigned)z < DIM);
    int lin = inb ? (x * DIM + y) * DIM + z : 0;
    bool v = inb && (occ[lin] != 0);
    valid_out[i] = v ? 1.0f : 0.0f;
    if (v) {
        const float4* __restrict__ src = (const float4*)row;
        float4* __restrict__ dst = (float4*)(gvol + (long)lin * CCH);
#pragma unroll
        for (int k = 0; k < CCH / 4; ++k) dst[k] = src[k];
    }
}

// ---------------------------------------------------------------------------
// Target pass 1: global targets (bounds-checked vs relative_origin).
// ---------------------------------------------------------------------------
__global__ void target_global_kernel(const int* __restrict__ gtr,
                                     const float* __restrict__ gtsdf,
                                     const int* __restrict__ origin,
                                     float* __restrict__ validt_out,
                                     float* __restrict__ tvol, int n) {
    int i = blockIdx.x * blockDim.x + threadIdx.x;
    if (i >= n) return;
    int x = gtr[3 * i + 0] - origin[0];
    int y = gtr[3 * i + 1] - origin[1];
    int z = gtr[3 * i + 2] - origin[2];
    bool inb = ((unsigned)x < DIM) && ((unsigned)y < DIM) && ((unsigned)z < DIM);
    validt_out[i] = inb ? 1.0f : 0.0f;
    if (inb) tvol[(x * DIM + y) * DIM + z] = gtsdf[i];
}

// ---------------------------------------------------------------------------
// Target pass 2: current-frame targets (always in range); launched after
// pass 1 so these win index collisions (matches concatenation order).
// ---------------------------------------------------------------------------
__global__ void target_current_kernel(const int* __restrict__ coords,
                                      const float* __restrict__ tsdf,
                                      float* __restrict__ tvol, int n) {
    int i = blockIdx.x * blockDim.x + threadIdx.x;
    if (i >= n) return;
    int x = coords[3 * i + 0];
    int y = coords[3 * i + 1];
    int z = coords[3 * i + 2];
    tvol[(x * DIM + y) * DIM + z] = tsdf[i];
}

extern "C" void kernel_launch(void* const* d_in, const int* in_sizes, int n_in,
                              void* d_out, int out_size, void* d_ws, size_t ws_size,
                              hipStream_t stream) {
    const int*   current_coords     = (const int*)  d_in[0];
    const float* current_values     = (const float*)d_in[1];
    const int*   global_coords_raw  = (const int*)  d_in[2];
    const float* global_values      = (const float*)d_in[3];
    const int*   coords_tgt_global  = (const int*)  d_in[4];
    const float* tsdf_target        = (const float*)d_in[5];
    const int*   global_tgt_raw     = (const int*)  d_in[6];
    const float* global_tsdf_target = (const float*)d_in[7];
    const int*   relative_origin    = (const int*)  d_in[8];

    const int n_cur  = in_sizes[0] / 3;   // 150000
    const int n_glob = in_sizes[2] / 3;   // 200000
    const int n_tcur = in_sizes[4] / 3;   // 100000
    const int n_tgl  = in_sizes[6] / 3;   // 120000

    const long VOL_FEAT = (long)DIM * DIM * DIM * CCH;   // 21,233,664 = 48*27*16384
    const long VOL_1    = (long)DIM * DIM * DIM;         //    884,736 =  2*27*16384

    float* out = (float*)d_out;
    float* updated_coords = out;                               // 450000
    float* current_volume = updated_coords + (long)in_sizes[0];
    float* global_volume  = current_volume + VOL_FEAT;
    float* target_volume  = global_volume + VOL_FEAT;
    float* valid_out      = target_volume + VOL_1;             // 200000
    float* validt_out     = valid_out + n_glob;                // 120000

    int* occ = (int*)d_ws;  // DIM^3 int32 occupancy volume (3.5 MB)

    const int T = 256;
    auto blocks = [](long n, int t) { return (int)((n + t - 1) / t); };

    // --- Phase 0: TDM bulk fills (DMA engine streams LDS tile -> HBM) ---
    tdm_fill_kernel<<<2, T, 0, stream>>>((float*)occ, 27, 0.0f);        // int 0 == fp 0.0
    tdm_fill_kernel<<<48, T, 0, stream>>>(current_volume, 27, 0.0f);
    tdm_fill_kernel<<<48, T, 0, stream>>>(global_volume, 27, 0.0f);
    tdm_fill_kernel<<<2, T, 0, stream>>>(target_volume, 27, 1.0f);
    i2f_kernel<<<blocks(in_sizes[0], T), T, 0, stream>>>(
        current_coords, updated_coords, in_sizes[0]);

    // --- Phase 1: occupancy scatter (must precede global pass) ---
    scatter_occ_kernel<<<blocks(n_cur, T), T, 0, stream>>>(current_coords, occ, n_cur);

    // --- Phase 2: feature scatters ---
    scatter_feat_kernel<<<blocks(n_cur, T), T, 0, stream>>>(
        current_coords, current_values, current_volume, n_cur);
    global_pass_kernel<<<blocks(n_glob, T), T, 0, stream>>>(
        global_coords_raw, global_values, relative_origin, occ,
        valid_out, global_volume, n_glob);

    // --- Phase 3: target volume (global targets first, current targets win ties) ---
    target_global_kernel<<<blocks(n_tgl, T), T, 0, stream>>>(
        global_tgt_raw, global_tsdf_target, relative_origin,
        validt_out, target_volume, n_tgl);
    target_current_kernel<<<blocks(n_tcur, T), T, 0, stream>>>(
        coords_tgt_global, tsdf_target, target_volume, n_tcur);
}